// EfficientGravNetLayer_55929064129395
// MI455X (gfx1250) — compile-verified
//
#include <hip/hip_runtime.h>
#include <hip/hip_bf16.h>

typedef float v2f __attribute__((ext_vector_type(2)));
typedef float v8f __attribute__((ext_vector_type(8)));

#define HCONST 2048
#define LOGH   11
#define FD     64
#define HD     64
#define KNBR   16

// ---------------- s = x @ Wc + bc ----------------
__global__ void s_kernel(const float* __restrict__ X, const float* __restrict__ Wc,
                         const float* __restrict__ bc, float* __restrict__ s, int N) {
    __shared__ float wsh[FD];
    if (threadIdx.x < FD) wsh[threadIdx.x] = Wc[threadIdx.x];
    __syncthreads();
    int i = blockIdx.x * blockDim.x + threadIdx.x;
    if (i < N) {
        float acc = bc[0];
        const float* xr = X + (long)i * FD;
        #pragma unroll
        for (int j = 0; j < FD; ++j) acc += xr[j] * wsh[j];
        s[i] = acc;
    }
}

// ---------------- W1 = Wu_top + Wr ; b2 = bu + br ----------------
__global__ void prep_kernel(const float* __restrict__ Wu, const float* __restrict__ bu,
                            const float* __restrict__ Wr, const float* __restrict__ br,
                            float* __restrict__ W1, float* __restrict__ b2) {
    int i = blockIdx.x * blockDim.x + threadIdx.x;
    if (i < FD * HD) W1[i] = Wu[i] + Wr[i];
    if (i < HD)      b2[i] = bu[i] + br[i];
}

// ---------------- per-batch bitonic sort (2048 elements) ----------------
__global__ void __launch_bounds__(1024)
sort_kernel(const float* __restrict__ s, float* __restrict__ s_sorted,
            int* __restrict__ sorted_idx) {
    __shared__ float key[HCONST];
    __shared__ int   idx[HCONST];
    const int base = blockIdx.x * HCONST;
    const int tid  = threadIdx.x;
    for (int t = tid; t < HCONST; t += 1024) { key[t] = s[base + t]; idx[t] = t; }
    __syncthreads();
    for (unsigned k = 2; k <= HCONST; k <<= 1) {
        for (unsigned j = k >> 1; j > 0; j >>= 1) {
            for (unsigned t = tid; t < HCONST; t += 1024) {
                unsigned ixj = t ^ j;
                if (ixj > t) {
                    bool up = ((t & k) == 0);
                    float a = key[t], c = key[ixj];
                    if ((a > c) == up) {
                        key[t] = c; key[ixj] = a;
                        int ti = idx[t]; idx[t] = idx[ixj]; idx[ixj] = ti;
                    }
                }
            }
            __syncthreads();
        }
    }
    for (int t = tid; t < HCONST; t += 1024) {
        s_sorted[base + t]   = key[t];
        sorted_idx[base + t] = base + idx[t];
    }
}

// ---------------- WMMA f32 16x16x4 GEMM helper (row-major A[M,64], W[64,64]) ----------------
__device__ inline v8f wmma_pass_f32(const float* __restrict__ A, const float* __restrict__ W,
                                    int m0, int n0, int lrow, int lhi, v8f acc) {
    #pragma unroll
    for (int k0 = 0; k0 < FD; k0 += 4) {
        int kk = k0 + 2 * lhi;               // lanes 0-15: K=k0,k0+1 ; lanes 16-31: K=k0+2,k0+3
        v2f a = *reinterpret_cast<const v2f*>(A + (long)(m0 + lrow) * FD + kk);
        v2f b;
        b.x = W[kk * HD + n0 + lrow];
        b.y = W[(kk + 1) * HD + n0 + lrow];
        acc = __builtin_amdgcn_wmma_f32_16x16x4_f32(false, a, false, b, (short)0, acc,
                                                    false, false);
    }
    return acc;
}

// ---------------- feat = x @ Wf + bf ----------------
__global__ void __launch_bounds__(256)
feat_gemm(const float* __restrict__ X, const float* __restrict__ Wf,
          const float* __restrict__ bf, float* __restrict__ feat) {
    int wave  = blockIdx.x * (blockDim.x >> 5) + (threadIdx.x >> 5);
    int lane  = threadIdx.x & 31;
    int m0    = (wave >> 2) * 16;
    int n0    = (wave & 3) * 16;
    int lrow  = lane & 15, lhi = lane >> 4;
    v8f acc = {};
    acc = wmma_pass_f32(X, Wf, m0, n0, lrow, lhi, acc);
    int col = n0 + lrow;
    float bias = bf[col];
    #pragma unroll
    for (int r = 0; r < 8; ++r) {
        int row = m0 + r + 8 * lhi;
        feat[(long)row * HD + col] = acc[r] + bias;
    }
}

// ---------------- neighbor aggregation (scatter to unsorted order) ----------------
__global__ void __launch_bounds__(256)
agg_kernel(const float* __restrict__ feat, const float* __restrict__ s_sorted,
           const int* __restrict__ sorted_idx, float* __restrict__ aggU, int N) {
    __shared__ float w_s[4 * KNBR];
    __shared__ int   g_s[4 * KNBR];
    const int tid = threadIdx.x;
    const int nodeBase = blockIdx.x * 4;
    if (tid < 4 * KNBR) {
        int node = tid >> 4;
        int k    = tid & 15;
        int i    = nodeBase + node;                 // sorted position
        int b    = i >> LOGH;
        int off  = (k < 8) ? (k - 8) : (k - 7);     // -8..-1, 1..8
        int nbr  = i + off;
        nbr = min(max(nbr, 0), N - 1);
        if ((nbr >> LOGH) != b) {                   // one-step fixup, exactly as reference
            int nv = (nbr > i) ? (nbr - 1) : (nbr + 1);
            nbr = min(max(nv, 0), N - 1);
        }
        float ds = s_sorted[i] - s_sorted[nbr];
        float w  = fmaxf(expf(-10.0f * ds * ds), 1e-6f);
        w_s[tid] = w;
        g_s[tid] = sorted_idx[nbr];                 // original-row index of neighbor
    }
    __syncthreads();
    int node = tid >> 6;
    int h    = tid & 63;
    int i    = nodeBase + node;
    float acc = 0.0f, wsum = 0.0f;
    #pragma unroll
    for (int k = 0; k < KNBR; ++k) {
        float w = w_s[node * KNBR + k];
        acc  += w * feat[(long)g_s[node * KNBR + k] * HD + h];
        wsum += w;
    }
    wsum = wsum + ((wsum == 0.0f) ? 1.0f : 0.0f);
    wsum = fmaxf(wsum, 1e-6f);
    aggU[(long)sorted_idx[i] * HD + h] = acc / wsum;
}

// ---------------- out = x@W1 + aggU@Wu_bot + b2, masked ----------------
__global__ void __launch_bounds__(256)
out_gemm(const float* __restrict__ X, const float* __restrict__ aggU,
         const float* __restrict__ W1, const float* __restrict__ Wu_bot,
         const float* __restrict__ b2, const unsigned char* __restrict__ mask,
         float* __restrict__ out) {
    int wave  = blockIdx.x * (blockDim.x >> 5) + (threadIdx.x >> 5);
    int lane  = threadIdx.x & 31;
    int m0    = (wave >> 2) * 16;
    int n0    = (wave & 3) * 16;
    int lrow  = lane & 15, lhi = lane >> 4;
    v8f acc = {};
    acc = wmma_pass_f32(X,    W1,     m0, n0, lrow, lhi, acc);
    acc = wmma_pass_f32(aggU, Wu_bot, m0, n0, lrow, lhi, acc);
    int col = n0 + lrow;
    float bias = b2[col];
    #pragma unroll
    for (int r = 0; r < 8; ++r) {
        int row = m0 + r + 8 * lhi;
        float mv = mask[row] ? 1.0f : 0.0f;
        out[(long)row * HD + col] = (acc[r] + bias) * mv;
    }
}

extern "C" void kernel_launch(void* const* d_in, const int* in_sizes, int n_in,
                              void* d_out, int out_size, void* d_ws, size_t ws_size,
                              hipStream_t stream) {
    const float* x    = (const float*)d_in[0];
    const unsigned char* mask = (const unsigned char*)d_in[1];
    const float* Wc   = (const float*)d_in[2];
    const float* bc   = (const float*)d_in[3];
    const float* Wf   = (const float*)d_in[4];
    const float* bf   = (const float*)d_in[5];
    const float* Wu   = (const float*)d_in[6];
    const float* bu   = (const float*)d_in[7];
    const float* Wr   = (const float*)d_in[8];
    const float* br   = (const float*)d_in[9];
    float* out = (float*)d_out;

    const int N = in_sizes[1];          // B*H = 65536
    float* ws = (float*)d_ws;
    float* s_buf    = ws;                               // N
    float* s_sorted = ws + N;                           // N
    int*   sidx     = (int*)(ws + 2 * (long)N);         // N ints
    float* W1       = ws + 3 * (long)N;                 // 4096
    float* b2       = W1 + FD * HD;                     // 64
    float* feat     = b2 + HD;                          // N*64
    float* aggU     = feat + (long)N * HD;              // N*64

    // 1. sort coordinate
    s_kernel<<<(N + 255) / 256, 256, 0, stream>>>(x, Wc, bc, s_buf, N);
    // 2. folded weights
    prep_kernel<<<(FD * HD + 255) / 256, 256, 0, stream>>>(Wu, bu, Wr, br, W1, b2);
    // 3. feat GEMM: (N/16)*(64/16) waves, 8 waves/block
    int tiles = (N / 16) * (HD / 16);
    feat_gemm<<<tiles / 8, 256, 0, stream>>>(x, Wf, bf, feat);
    // 4. per-batch bitonic sort
    sort_kernel<<<N / HCONST, 1024, 0, stream>>>(s_buf, s_sorted, sidx);
    // 5. neighbor aggregation, scattered back to original order
    agg_kernel<<<N / 4, 256, 0, stream>>>(feat, s_sorted, sidx, aggU, N);
    // 6. output GEMM (x·W1 + aggU·Wu_bot + b2), masked
    out_gemm<<<tiles / 8, 256, 0, stream>>>(x, aggU, W1, Wu + FD * HD, b2, mask, out);
}